// FraudDetectionHybridModel_65481071406852
// MI455X (gfx1250) — compile-verified
//
#include <hip/hip_runtime.h>
#include <stdint.h>

typedef __attribute__((ext_vector_type(4))) unsigned int u32x4;
typedef __attribute__((ext_vector_type(8))) unsigned int u32x8;

#define NL 5
#define QD 2
#define BLOCK 256
#define SPT 4                                   // samples per thread
#define SAMPLES_PER_BLOCK (BLOCK * SPT)         // 1024
#define FLOATS_PER_BLOCK (SAMPLES_PER_BLOCK*2)  // 2048 floats = 8 KB LDS
#define FLOATS_PER_WAVE (32 * SPT * 2)          // 256 floats = 1 KB per-wave TDM tile

struct Params {
  float W[NL][2][2], b[NL][2], sc[NL][2], sh[NL][2];
  float wf0, wf1, bf0;
  float ct[QD][2], st[QD][2];
};

// Prefer the gfx1250 hardware tanh (V_TANH_F32, single trans op).
// Fallback: tanh(v) = 1 - 2/(exp(2v)+1) -> v_exp_f32 + v_rcp_f32 (+3 VALU);
// saturates correctly to +/-1 for large |v| (exp->inf, rcp->0).
__device__ __forceinline__ float fast_tanh(float v) {
#if __has_builtin(__builtin_amdgcn_tanhf)
  return __builtin_amdgcn_tanhf(v);
#else
  float e = __builtin_amdgcn_exp2f(v * 2.8853900817779268f); // 2*log2(e)
  float r = __builtin_amdgcn_rcpf(e + 1.0f);
  return fmaf(-2.0f, r, 1.0f);
#endif
}

// Givens rotation (a,b) <- (c*a - s*b, s*a + c*b)
__device__ __forceinline__ void rot(float& a, float& b, float c, float s) {
  float na = fmaf(c, a, -(s * b));
  float nb = fmaf(s, a,  (c * b));
  a = na; b = nb;
}

__device__ __forceinline__ void load_params(Params& p,
    const float* __restrict__ Ws, const float* __restrict__ bs,
    const float* __restrict__ scales, const float* __restrict__ shifts,
    const float* __restrict__ Wf, const float* __restrict__ bf,
    const float* __restrict__ theta) {
#pragma unroll
  for (int l = 0; l < NL; ++l) {
#pragma unroll
    for (int j = 0; j < 2; ++j) {
      p.W[l][j][0] = Ws[l*4 + j*2 + 0];
      p.W[l][j][1] = Ws[l*4 + j*2 + 1];
      p.b[l][j]    = bs[l*2 + j];
      p.sc[l][j]   = scales[l*2 + j];
      p.sh[l][j]   = shifts[l*2 + j];
    }
  }
  p.wf0 = Wf[0]; p.wf1 = Wf[1]; p.bf0 = bf[0];
#pragma unroll
  for (int d = 0; d < QD; ++d) {
#pragma unroll
    for (int q = 0; q < 2; ++q) {
      float a = 0.5f * theta[d*2 + q];
      p.ct[d][q] = __cosf(a);
      p.st[d][q] = __sinf(a);
    }
  }
}

__device__ __forceinline__ float eval_sample(float x0, float x1, const Params& p) {
  // ---- quantum head: 2-qubit statevector, real(R)/imag(I) 4-vectors ----
  float c0 = __cosf(0.5f * x0), s0 = __sinf(0.5f * x0);
  float c1 = __cosf(0.5f * x1), s1 = __sinf(0.5f * x1);
  // RX(x0) (x) RX(x1) |00>:  amp = (c1,-i s1) (x) (c0,-i s0)
  float R0 =  c1 * c0, R1 = 0.f,       R2 = 0.f,       R3 = -s1 * s0;
  float I0 =  0.f,     I1 = -c1 * s0,  I2 = -s1 * c0,  I3 = 0.f;
#pragma unroll
  for (int d = 0; d < QD; ++d) {
    float c = p.ct[d][0], s = p.st[d][0];        // RY on qubit0: pairs (0,1),(2,3)
    rot(R0, R1, c, s); rot(R2, R3, c, s);
    rot(I0, I1, c, s); rot(I2, I3, c, s);
    c = p.ct[d][1]; s = p.st[d][1];              // RY on qubit1: pairs (0,2),(1,3)
    rot(R0, R2, c, s); rot(R1, R3, c, s);
    rot(I0, I2, c, s); rot(I1, I3, c, s);
    R3 = -R3; I3 = -I3;                          // CZ: negate |11> (free src-neg)
  }
  // <Z0>+<Z1> = [2,0,0,-2] . probs
  float q = 2.0f * ((R0*R0 + I0*I0) - (R3*R3 + I3*I3));

  // ---- classical backbone ----
  float h0 = x0, h1 = x1;
#pragma unroll
  for (int l = 0; l < NL; ++l) {
    float a0 = fmaf(h1, p.W[l][0][1], fmaf(h0, p.W[l][0][0], p.b[l][0]));
    float a1 = fmaf(h1, p.W[l][1][1], fmaf(h0, p.W[l][1][0], p.b[l][1]));
    h0 = fmaf(fast_tanh(a0), p.sc[l][0], p.sh[l][0]);
    h1 = fmaf(fast_tanh(a1), p.sc[l][1], p.sh[l][1]);
  }
  float cls = fmaf(h1, p.wf1, fmaf(h0, p.wf0, p.bf0));
  return cls + q;
}

__global__ __launch_bounds__(BLOCK) void fraud_hybrid_kernel(
    const float* __restrict__ x,  const float* __restrict__ Ws,
    const float* __restrict__ bs, const float* __restrict__ scales,
    const float* __restrict__ shifts, const float* __restrict__ Wf,
    const float* __restrict__ bf, const float* __restrict__ theta,
    float* __restrict__ out, int n) {
  __shared__ float tile[FLOATS_PER_BLOCK];

  const int tid  = threadIdx.x;
  const int wave = tid >> 5;
  const int lane = tid & 31;
  const long long blockBase = (long long)blockIdx.x * SAMPLES_PER_BLOCK;

  Params p;
  load_params(p, Ws, bs, scales, shifts, Wf, bf, theta);

  if (blockBase + SAMPLES_PER_BLOCK <= (long long)n) {
    // ---- fast path: per-wave TDM DMA of this wave's 1KB x-slice into LDS ----
    unsigned lds_off = (unsigned)(uintptr_t)&tile[wave * FLOATS_PER_WAVE];
    uint64_t gaddr = (uint64_t)(uintptr_t)x +
        ((uint64_t)blockBase * 2ull + (uint64_t)(wave * FLOATS_PER_WAVE)) * 4ull;
    unsigned g_lo = (unsigned)__builtin_amdgcn_readfirstlane((int)(unsigned)gaddr);
    unsigned g_hi = (unsigned)__builtin_amdgcn_readfirstlane((int)(unsigned)(gaddr >> 32));
    lds_off       = (unsigned)__builtin_amdgcn_readfirstlane((int)lds_off);

    // D# group0: count=1 | lds_addr | global_addr[56:0] | type=2
    u32x4 d0 = { 1u, lds_off, g_lo, (g_hi & 0x01FFFFFFu) | (2u << 30) };
    // D# group1: mask=0, data_size=4B, tensor_dim0=tile_dim0=256 elems,
    //            tensor_dim1=tile_dim1=1, tensor_dim0_stride=256
    u32x8 d1 = { 0x00020000u,
                 ((unsigned)FLOATS_PER_WAVE & 0xFFFFu) << 16,
                 0x00010000u,
                 ((unsigned)FLOATS_PER_WAVE & 0xFFFFu) << 16,
                 1u,
                 (unsigned)FLOATS_PER_WAVE,
                 0u, 0u };
    asm volatile("tensor_load_to_lds %0, %1" :: "s"(d0), "s"(d1) : "memory");
    __builtin_amdgcn_s_wait_tensorcnt(0);
    asm volatile("" ::: "memory");

    const float4* src =
        (const float4*)&tile[wave * FLOATS_PER_WAVE + lane * (SPT * 2)];
    float4 va = src[0];   // samples s, s+1
    float4 vb = src[1];   // samples s+2, s+3

    float4 r;
    r.x = eval_sample(va.x, va.y, p);
    r.y = eval_sample(va.z, va.w, p);
    r.z = eval_sample(vb.x, vb.y, p);
    r.w = eval_sample(vb.z, vb.w, p);

    long long s0i = blockBase + (long long)(wave * 32 * SPT + lane * SPT);
    *(float4*)(out + s0i) = r;
  } else {
    // ---- tail path: direct global loads with bounds checks ----
    long long s = blockBase + (long long)tid * SPT;
#pragma unroll
    for (int k = 0; k < SPT; ++k) {
      long long i = s + k;
      if (i < (long long)n)
        out[i] = eval_sample(x[2*i], x[2*i + 1], p);
    }
  }
}

extern "C" void kernel_launch(void* const* d_in, const int* in_sizes, int n_in,
                              void* d_out, int out_size, void* d_ws, size_t ws_size,
                              hipStream_t stream) {
  const float* x      = (const float*)d_in[0];
  const float* Ws     = (const float*)d_in[1];
  const float* bs     = (const float*)d_in[2];
  const float* scales = (const float*)d_in[3];
  const float* shifts = (const float*)d_in[4];
  const float* Wf     = (const float*)d_in[5];
  const float* bf     = (const float*)d_in[6];
  const float* theta  = (const float*)d_in[7];
  float* out = (float*)d_out;

  int n = in_sizes[0] / 2;
  int blocks = (n + SAMPLES_PER_BLOCK - 1) / SAMPLES_PER_BLOCK;
  hipLaunchKernelGGL(fraud_hybrid_kernel, dim3(blocks), dim3(BLOCK), 0, stream,
                     x, Ws, bs, scales, shifts, Wf, bf, theta, out, n);
}